// MoERouter_70531952935128
// MI455X (gfx1250) — compile-verified
//
#include <hip/hip_runtime.h>
#include <math.h>

#define B_TOTAL 4096
#define NEXPERT 64
#define HDIM    512
#define TOPK    8
#define CTX_F   68          // x_context feature count; 68 = 17 * 4 (exact K=4 WMMA chunks)
#define NCHUNK  (CTX_F / 4) // 17
#define ROWS_PER_WAVE  16
#define WAVES_PER_BLOCK 2
#define ROWS_PER_BLOCK (ROWS_PER_WAVE * WAVES_PER_BLOCK)
#define THREADS (32 * WAVES_PER_BLOCK)
#define NPROJ (CTX_F + 7 + 1)   // 68 Wc_proj + 7 We_proj + 1 fused constant = 76

typedef __attribute__((ext_vector_type(2))) float v2f;
typedef __attribute__((ext_vector_type(8))) float v8f;

// ---------------------------------------------------------------------------
// Kernel 1: fold Ws into the weight matrices (linearity lets us pre-reduce H).
//   ws[0..67]  = Wc_proj[c] = sum_h Wc[c,h] * Ws[H+h]
//   ws[68..74] = We_proj[f] = sum_h We[f,h] * Ws[h]
//   ws[75]     = sum_h be[h]*Ws[h] + sum_h bc[h]*Ws[H+h] + bs
// ---------------------------------------------------------------------------
__global__ __launch_bounds__(128) void router_precompute(
    const float* __restrict__ Wc, const float* __restrict__ bc,
    const float* __restrict__ We, const float* __restrict__ be,
    const float* __restrict__ Ws, const float* __restrict__ bs,
    float* __restrict__ ws)
{
    int t = threadIdx.x;
    if (t < CTX_F) {
        const float* row = Wc + (size_t)t * HDIM;
        const float* wsc = Ws + HDIM;
        float acc = 0.f;
        for (int h = 0; h < HDIM; ++h) acc = fmaf(row[h], wsc[h], acc);
        ws[t] = acc;
    } else if (t < CTX_F + 7) {
        const float* row = We + (size_t)(t - CTX_F) * HDIM;
        float acc = 0.f;
        for (int h = 0; h < HDIM; ++h) acc = fmaf(row[h], Ws[h], acc);
        ws[t] = acc;
    } else if (t == CTX_F + 7) {
        float acc = bs[0];
        for (int h = 0; h < HDIM; ++h) acc = fmaf(be[h], Ws[h], acc);
        for (int h = 0; h < HDIM; ++h) acc = fmaf(bc[h], Ws[HDIM + h], acc);
        ws[t] = acc;
    }
}

// ---------------------------------------------------------------------------
// Kernel 2: fully fused router. One wave owns 16 batch rows.
// Phase 1: context dot via V_WMMA_F32_16X16X4_F32, 17 K-steps (K=68),
//          Wc_proj in B column 0, D[:,0] = the 16 context dots.
//          All A/B operands preloaded into registers, then a clean WMMA chain.
// Phase 2: per row, lane = 2 experts; logits, wave top-8, softmax, action out.
// ---------------------------------------------------------------------------
__global__ __launch_bounds__(THREADS) void router_main(
    const float* __restrict__ xc, const float* __restrict__ xq,
    const float* __restrict__ xr, const float* __restrict__ xk,
    const float* __restrict__ ws,
    float* __restrict__ out_act, float* __restrict__ out_gate,
    float* __restrict__ out_logit)
{
    __shared__ float wproj[NPROJ];
    __shared__ float ctx_s[ROWS_PER_BLOCK];

    const int tid  = threadIdx.x;
    const int lane = tid & 31;
    const int wave = tid >> 5;
    for (int i = tid; i < NPROJ; i += THREADS) wproj[i] = ws[i];  // NPROJ > THREADS
    __syncthreads();

    const int rowbase = blockIdx.x * ROWS_PER_BLOCK + wave * ROWS_PER_WAVE;

    // ---- Phase 1: WMMA context projection --------------------------------
    // A layout (f32 16x4): lanes 0-15 hold M=lane, VGPR0/1 = K0/K1;
    //                      lanes 16-31 hold M=lane-16, VGPR0/1 = K2/K3.
    // B (4x16): lane n holds B[khalf+j, n] in VGPR j; only column n==0 used.
    const int m     = lane & 15;
    const int khalf = (lane >> 4) * 2;           // 0 or 2
    const float* arow = xc + (size_t)(rowbase + m) * CTX_F;
    const float bsel = (m == 0) ? 1.0f : 0.0f;   // select column 0 by multiply (no divergence)

    float av0[NCHUNK], av1[NCHUNK], bv0[NCHUNK], bv1[NCHUNK];
#pragma unroll
    for (int c = 0; c < NCHUNK; ++c) {           // 17x global_load_b64, issued as a burst
        const int k0 = 4 * c + khalf;
        av0[c] = arow[k0];
        av1[c] = arow[k0 + 1];
    }
#pragma unroll
    for (int c = 0; c < NCHUNK; ++c) {           // broadcast LDS reads, unconditional
        const int k0 = 4 * c + khalf;
        bv0[c] = wproj[k0]     * bsel;
        bv1[c] = wproj[k0 + 1] * bsel;
    }

    v8f acc = {};
#pragma unroll
    for (int c = 0; c < NCHUNK; ++c) {
        v2f a, b;
        a.x = av0[c]; a.y = av1[c];
        b.x = bv0[c]; b.y = bv1[c];
        acc = __builtin_amdgcn_wmma_f32_16x16x4_f32(
            false, a, false, b, (short)0, acc, false, false);
    }

    const float cconst = wproj[CTX_F + 7];
    if (lane == 0) {
#pragma unroll
        for (int i = 0; i < 8; ++i) ctx_s[wave * ROWS_PER_WAVE + i] = acc[i] + cconst;
    } else if (lane == 16) {
#pragma unroll
        for (int i = 0; i < 8; ++i) ctx_s[wave * ROWS_PER_WAVE + 8 + i] = acc[i] + cconst;
    }
    __syncthreads();

    float wp[7];
#pragma unroll
    for (int f = 0; f < 7; ++f) wp[f] = wproj[CTX_F + f];

    // ---- Phase 2: logits, top-8, softmax, action logits ------------------
    for (int r = 0; r < ROWS_PER_WAVE; ++r) {
        const int row = rowbase + r;
        const float ctx = ctx_s[wave * ROWS_PER_WAVE + r];
        const int e0 = lane, e1 = lane + 32;
        const size_t be0 = (size_t)row * NEXPERT + e0;
        const size_t be1 = (size_t)row * NEXPERT + e1;

        const float* q0 = xq + be0 * 3;
        const float* q1 = xq + be1 * 3;
        const float* r0 = xr + be0 * 2;
        const float* r1 = xr + be1 * 2;
        const float* k0 = xk + be0 * 2;
        const float* k1 = xk + be1 * 2;

        const float q00 = q0[0], q01 = q0[1], q02 = q0[2];
        const float q10 = q1[0], q11 = q1[1], q12 = q1[2];

        const float l0 = ctx + q00 * wp[0] + q01 * wp[1] + q02 * wp[2]
                             + r0[0] * wp[3] + r0[1] * wp[4]
                             + k0[0] * wp[5] + k0[1] * wp[6];
        const float l1 = ctx + q10 * wp[0] + q11 * wp[1] + q12 * wp[2]
                             + r1[0] * wp[3] + r1[1] * wp[4]
                             + k1[0] * wp[5] + k1[1] * wp[6];

        out_logit[be0] = l0;
        out_logit[be1] = l1;

        // top-8 of 64 across the wave (value desc, index asc on ties)
        float a0 = l0, a1 = l1;
        float tv[TOPK];
        int   ti[TOPK];
#pragma unroll
        for (int i = 0; i < TOPK; ++i) {
            float best; int bidx;
            if (a0 >= a1) { best = a0; bidx = e0; }
            else          { best = a1; bidx = e1; }
#pragma unroll
            for (int off = 16; off >= 1; off >>= 1) {
                float ov = __shfl_xor(best, off, 32);
                int   oi = __shfl_xor(bidx, off, 32);
                if (ov > best || (ov == best && oi < bidx)) { best = ov; bidx = oi; }
            }
            tv[i] = best; ti[i] = bidx;
            if      (bidx == e0) a0 = -INFINITY;
            else if (bidx == e1) a1 = -INFINITY;
        }

        const float mx = tv[0];                 // first pick is the max
        float ssum = 0.f;
#pragma unroll
        for (int i = 0; i < TOPK; ++i) ssum += expf(tv[i] - mx);
        const float inv = 1.f / ssum;

        bool s0 = false, s1 = false;
#pragma unroll
        for (int i = 0; i < TOPK; ++i) { s0 |= (ti[i] == e0); s1 |= (ti[i] == e1); }
        const float g0 = s0 ? expf(l0 - mx) * inv : 0.f;   // exp(NEG_INF-max) -> 0
        const float g1 = s1 ? expf(l1 - mx) * inv : 0.f;

        out_gate[be0] = g0;
        out_gate[be1] = g1;

        float p0 = g0 * q00 + g1 * q10;
        float p1 = g0 * q01 + g1 * q11;
        float p2 = g0 * q02 + g1 * q12;
#pragma unroll
        for (int off = 16; off >= 1; off >>= 1) {
            p0 += __shfl_xor(p0, off, 32);
            p1 += __shfl_xor(p1, off, 32);
            p2 += __shfl_xor(p2, off, 32);
        }
        if (lane == 0) {
            out_act[(size_t)row * 3 + 0] = p0;
            out_act[(size_t)row * 3 + 1] = p1;
            out_act[(size_t)row * 3 + 2] = p2;
        }
    }
}

extern "C" void kernel_launch(void* const* d_in, const int* in_sizes, int n_in,
                              void* d_out, int out_size, void* d_ws, size_t ws_size,
                              hipStream_t stream) {
    const float* xc = (const float*)d_in[0];
    const float* xq = (const float*)d_in[1];
    const float* xr = (const float*)d_in[2];
    const float* xk = (const float*)d_in[3];
    const float* Wc = (const float*)d_in[4];
    const float* bc = (const float*)d_in[5];
    const float* We = (const float*)d_in[6];
    const float* be = (const float*)d_in[7];
    const float* Ws = (const float*)d_in[8];
    const float* bs = (const float*)d_in[9];

    float* ws  = (float*)d_ws;
    float* out = (float*)d_out;
    float* out_act   = out;                                   // (4096, 3)
    float* out_gate  = out + (size_t)B_TOTAL * 3;             // (4096, 64)
    float* out_logit = out_gate + (size_t)B_TOTAL * NEXPERT;  // (4096, 64)

    router_precompute<<<1, 128, 0, stream>>>(Wc, bc, We, be, Ws, bs, ws);
    router_main<<<B_TOTAL / ROWS_PER_BLOCK, THREADS, 0, stream>>>(
        xc, xq, xr, xk, ws, out_act, out_gate, out_logit);
}